// _QuantumGate_65481071407515
// MI455X (gfx1250) — compile-verified
//
#include <hip/hip_runtime.h>
#include <hip/hip_bf16.h>

// Quantum circuit simulation, restructured for MI455X (gfx1250):
//  - initial RX layer -> direct product-state construction (no gate sweep)
//  - per layer: unitary part = A3(x)A2(x)A1(x)A0 (nibble-fused 16x16 complex,
//    shared across batch), CNOT chain = GF(2) suffix-xor permutation applied
//    in the store addressing of PassB.
//  - 16x16 complex gate == 32x32 real matrix (interleaved re/im embedding),
//    applied to LDS tiles via chained V_WMMA_F32_16X16X4_F32 (fp32 == the
//    natural precision for complex64).
//  - LDS tiles padded 2 complexes / 16 to kill the 16-way bank conflict of the
//    nibble-0 gate (row stride 36 dwords, 9 coprime 16 -> 16 distinct banks).
//  - global->LDS staging via GLOBAL_LOAD_ASYNC_TO_LDS_B128 when the builtin
//    exists (ASYNCcnt path), plain vector copy otherwise.
//  - deterministic two-stage reduction for <Z> (no float atomics).
// ws layout: [ gates 64KB | partials 512KB | pad | state0 256MB | state1 256MB ]
// (needs >= 513MB of scratch)

#define NQ 16
#define NSTATE 65536
#define QBATCH 512
#define QDEPTH 4

typedef float v8f __attribute__((ext_vector_type(8)));
typedef float v2f __attribute__((ext_vector_type(2)));

#if __has_builtin(__builtin_amdgcn_global_load_async_to_lds_b128) && \
    __has_builtin(__builtin_amdgcn_s_wait_asynccnt)
#define QC_ASYNC 1
typedef int qc_iv4 __attribute__((vector_size(4 * sizeof(int))));
__device__ __forceinline__ void qc_async_ld16(const void* g, void* l)
{
    __builtin_amdgcn_global_load_async_to_lds_b128(
        (qc_iv4*)g, (qc_iv4*)l, 0, 0);
}
#endif

// ---------------------------------------------------------------------------
// Build the 16 fused nibble gates (layer x nibble), real-embedded 32x32.
// Qubit q of the reference owns bit (15-q) of the state index.
// ---------------------------------------------------------------------------
__global__ void qc_build_gates(const float* __restrict__ params,
                               float* __restrict__ gates)
{
    int t = threadIdx.x;
    if (t >= QDEPTH * 4) return;
    int L = t >> 2;   // layer
    int g = t & 3;    // bit-nibble (bits 4g..4g+3)

    float Mr[16][16], Mi[16][16];
    Mr[0][0] = 1.f; Mi[0][0] = 0.f;
    int dim = 1;
    for (int j = 0; j < 4; ++j) {
        int q = 15 - (4 * g + j);              // qubit owning bit 4g+j
        const float* p = params + (L * NQ + q) * 3;
        float cz = cosf(0.5f * p[0]), sz = sinf(0.5f * p[0]);
        float cx = cosf(0.5f * p[1]), sx = sinf(0.5f * p[1]);
        float cy = cosf(0.5f * p[2]), sy = sinf(0.5f * p[2]);
        // Rz = [[cz - i sz, 0],[0, cz + i sz]]
        float rzr[2][2] = {{cz, 0.f}, {0.f, cz}};
        float rzi[2][2] = {{-sz, 0.f}, {0.f, sz}};
        // Rx = [[cx, -i sx],[-i sx, cx]]
        float rxr[2][2] = {{cx, 0.f}, {0.f, cx}};
        float rxi[2][2] = {{0.f, -sx}, {-sx, 0.f}};
        // Ry = [[cy, -sy],[sy, cy]]
        float ryr[2][2] = {{cy, -sy}, {sy, cy}};
        float ryi[2][2] = {{0.f, 0.f}, {0.f, 0.f}};
        // T = Rx*Rz ; U = Ry*T
        float tr[2][2], ti[2][2], ur[2][2], ui[2][2];
        for (int a = 0; a < 2; ++a)
            for (int b = 0; b < 2; ++b) {
                float rr = 0.f, ii = 0.f;
                for (int k = 0; k < 2; ++k) {
                    rr += rxr[a][k] * rzr[k][b] - rxi[a][k] * rzi[k][b];
                    ii += rxr[a][k] * rzi[k][b] + rxi[a][k] * rzr[k][b];
                }
                tr[a][b] = rr; ti[a][b] = ii;
            }
        for (int a = 0; a < 2; ++a)
            for (int b = 0; b < 2; ++b) {
                float rr = 0.f, ii = 0.f;
                for (int k = 0; k < 2; ++k) {
                    rr += ryr[a][k] * tr[k][b] - ryi[a][k] * ti[k][b];
                    ii += ryr[a][k] * ti[k][b] + ryi[a][k] * tr[k][b];
                }
                ur[a][b] = rr; ui[a][b] = ii;
            }
        // M = kron(U, M): bit j becomes the new high bit. Blocks with a|b!=0
        // write outside the old [0,dim)^2 region, so write (0,0) block last.
        for (int a = 1; a >= 0; --a)
            for (int b = 1; b >= 0; --b)
                for (int r = dim - 1; r >= 0; --r)
                    for (int c = dim - 1; c >= 0; --c) {
                        float mr = Mr[r][c], mi = Mi[r][c];
                        Mr[a * dim + r][b * dim + c] = ur[a][b] * mr - ui[a][b] * mi;
                        Mi[a * dim + r][b * dim + c] = ur[a][b] * mi + ui[a][b] * mr;
                    }
        dim <<= 1;
    }
    // Real embedding, interleaved re/im: rows/cols 2i,2i+1.
    float* G = gates + (size_t)(L * 4 + g) * 1024;
    for (int r = 0; r < 16; ++r)
        for (int c = 0; c < 16; ++c) {
            G[(2 * r) * 32 + 2 * c]         =  Mr[r][c];
            G[(2 * r) * 32 + 2 * c + 1]     = -Mi[r][c];
            G[(2 * r + 1) * 32 + 2 * c]     =  Mi[r][c];
            G[(2 * r + 1) * 32 + 2 * c + 1] =  Mr[r][c];
        }
}

// ---------------------------------------------------------------------------
// Product state after the initial RX layer: amp(b) = (-i)^pc * prod factors.
// ---------------------------------------------------------------------------
__global__ void __launch_bounds__(256) qc_init_state(const float* __restrict__ x,
                                                     float2* __restrict__ state)
{
    int h = blockIdx.x;   // bits 12..15
    int b = blockIdx.y;   // batch
    __shared__ float cs[NQ], sn[NQ];
    if (threadIdx.x < NQ) {
        float a = 0.5f * x[b * NQ + threadIdx.x];
        cs[threadIdx.x] = cosf(a);
        sn[threadIdx.x] = sinf(a);
    }
    __syncthreads();
    float2* dst = state + (size_t)b * NSTATE;
    for (int v = 0; v < 16; ++v) {
        int idx = (h << 12) | (v << 8) | (int)threadIdx.x;
        float mag = 1.f;
        int pc = 0;
        #pragma unroll
        for (int j = 0; j < NQ; ++j) {
            int bit = (idx >> j) & 1;
            int q = 15 - j;                         // qubit owning bit j
            mag *= bit ? sn[q] : cs[q];
            pc += bit;
        }
        float re = 0.f, im = 0.f;
        switch (pc & 3) {
            case 0: re =  mag; break;
            case 1: im = -mag; break;
            case 2: re = -mag; break;
            case 3: im =  mag; break;
        }
        dst[idx] = make_float2(re, im);
    }
}

// ---------------------------------------------------------------------------
// Padded LDS tile: physical(l) = l + 2*(l>>4)  (2 pad complexes per 16).
// 4096 logical complexes -> 4608 slots (36 KB). float4 view: f4(q)=q+(q>>3).
// ---------------------------------------------------------------------------
#define QC_TILE 4608

__device__ __forceinline__ int qc_phys(int l) { return l + ((l >> 4) << 1); }

// nibble value k at bit `shift` of the 12-bit local index; c = other 8 bits.
__device__ __forceinline__ int qc_lidx(int k, int c, int shift)
{
    int lo = c & ((1 << shift) - 1);
    int hi = c & ~((1 << shift) - 1);
    return qc_phys((hi << 4) | (k << shift) | lo);
}
__device__ __forceinline__ float qc_ldx(const float2* t, int r, int c, int shift)
{
    const float* f = (const float*)&t[qc_lidx(r >> 1, c, shift)];
    return f[r & 1];
}
__device__ __forceinline__ void qc_sty(float2* t, int r, int c, int shift, float v)
{
    float* f = (float*)&t[qc_lidx(r >> 1, c, shift)];
    f[r & 1] = v;
}

// Y(32x256) = G(32x32) * X(32x256) on one LDS tile; 256-thread block
// cooperates (8 waves x 2 column tiles x 2 row halves x 8 chained K=4 WMMAs).
// Fully unrolled/uniform control flow: EXEC stays all-ones around every WMMA.
__device__ __forceinline__ void qc_apply_gate(const float* __restrict__ G,
                                              const float2* __restrict__ src,
                                              float2* __restrict__ dst, int shift)
{
    int lane = threadIdx.x & 31;
    int wave = threadIdx.x >> 5;
    int hi16 = lane >> 4;    // 0: lanes 0-15, 1: lanes 16-31
    int l16  = lane & 15;
    #pragma unroll
    for (int half = 0; half < 2; ++half) {         // 2 column tiles per wave
        int tt = wave + half * 8;
        int col = tt * 16 + l16;
        #pragma unroll
        for (int i = 0; i < 2; ++i) {              // output row halves
            v8f acc = {};
            #pragma unroll
            for (int kb = 0; kb < 8; ++kb) {       // K = 32 in chunks of 4
                int kbase = kb * 4;
                v2f a, bf;
                // A frag (16x4 f32): VGPR0 = K kbase+0 / kbase+2 per half-wave
                a.x = G[(i * 16 + l16) * 32 + kbase + 2 * hi16];
                a.y = G[(i * 16 + l16) * 32 + kbase + 2 * hi16 + 1];
                // B frag (4x16 f32): rows striped across lanes per VGPR
                bf.x = qc_ldx(src, kbase + hi16,     col, shift);
                bf.y = qc_ldx(src, kbase + 2 + hi16, col, shift);
                acc = __builtin_amdgcn_wmma_f32_16x16x4_f32(
                          false, a, false, bf, (short)0, acc, false, false);
            }
            #pragma unroll
            for (int r = 0; r < 8; ++r)            // D: VGPR r -> M=r / r+8
                qc_sty(dst, i * 16 + r + 8 * hi16, col, shift, acc[r]);
        }
    }
}

// ---------------------------------------------------------------------------
// PassA: apply nibble gates 0,1,2 to a contiguous 4096-complex slice. In-place.
// ---------------------------------------------------------------------------
__global__ void __launch_bounds__(256) qc_passA(float2* __restrict__ state,
                                                const float* __restrict__ gates)
{
    __shared__ float2 tA[QC_TILE];
    __shared__ float2 tB[QC_TILE];
    int h = blockIdx.x, b = blockIdx.y;
    float2* S = state + (size_t)b * NSTATE + ((size_t)h << 12);
    const float4* gsrc = (const float4*)S;
    float4* la = (float4*)tA;
    for (int q = threadIdx.x; q < 2048; q += 256) {
        int qp = q + (q >> 3);                     // padded float4 slot
#ifdef QC_ASYNC
        qc_async_ld16(gsrc + q, la + qp);
#else
        la[qp] = gsrc[q];
#endif
    }
#ifdef QC_ASYNC
    __builtin_amdgcn_s_wait_asynccnt(0);
#endif
    __syncthreads();
    qc_apply_gate(gates + 0 * 1024, tA, tB, 0);
    __syncthreads();
    qc_apply_gate(gates + 1 * 1024, tB, tA, 4);
    __syncthreads();
    qc_apply_gate(gates + 2 * 1024, tA, tB, 8);
    __syncthreads();
    float4* gdst = (float4*)S;
    const float4* lb = (const float4*)tB;
    for (int q = threadIdx.x; q < 2048; q += 256) gdst[q] = lb[q + (q >> 3)];
}

// ---------------------------------------------------------------------------
// PassB: apply nibble-3 gate across stride-4096 rows, then scatter through the
// CNOT-chain permutation (GF(2) suffix-xor) into the other buffer.
// ---------------------------------------------------------------------------
__global__ void __launch_bounds__(256) qc_passB(const float2* __restrict__ src,
                                                float2* __restrict__ dst,
                                                const float* __restrict__ G3)
{
    __shared__ float2 tA[QC_TILE];
    __shared__ float2 tB[QC_TILE];
    int w = blockIdx.x, b = blockIdx.y;    // low12 = w*256 + c
    const float2* Sb = src + (size_t)b * NSTATE;
    float4* la = (float4*)tA;
    for (int q = threadIdx.x; q < 2048; q += 256) {
        int k = q >> 7, cp = q & 127;      // row k (bits 12-15), float4 pair
        const float4* grow = (const float4*)(Sb + k * 4096 + w * 256);
        int qp = q + (q >> 3);
#ifdef QC_ASYNC
        qc_async_ld16(grow + cp, la + qp);
#else
        la[qp] = grow[cp];
#endif
    }
#ifdef QC_ASYNC
    __builtin_amdgcn_s_wait_asynccnt(0);
#endif
    __syncthreads();
    qc_apply_gate(G3, tA, tB, 8);          // tile layout: [k][c] = k*256+c
    __syncthreads();
    float2* Db = dst + (size_t)b * NSTATE;
    for (int e = threadIdx.x; e < 4096; e += 256) {
        int k = e >> 8, c = e & 255;
        unsigned idx = ((unsigned)k << 12) | ((unsigned)w << 8) | (unsigned)c;
        unsigned t = idx;
        t ^= t >> 1; t ^= t >> 2; t ^= t >> 4; t ^= t >> 8;   // suffix-xor
        unsigned y = (t & 0x7FFFu) |
                     ((((t & 1u) ^ (idx >> 15)) & 1u) << 15); // wrap CNOT(15,0)
        Db[y] = tB[qc_phys(e)];
    }
}

// ---------------------------------------------------------------------------
// Per-qubit <Z> partials (deterministic; no float atomics):
//   part[b][h][q] = sum over slice h of p(b,idx) * (1 - 2*bit_{15-q}(idx))
// ---------------------------------------------------------------------------
__global__ void __launch_bounds__(256) qc_expz(const float2* __restrict__ state,
                                               float* __restrict__ part)
{
    int h = blockIdx.x, b = blockIdx.y;
    const float2* S = state + (size_t)b * NSTATE + ((size_t)h << 12);
    float acc[NQ];
    #pragma unroll
    for (int i = 0; i < NQ; ++i) acc[i] = 0.f;
    for (int v = 0; v < 16; ++v) {
        int loc = (v << 8) | (int)threadIdx.x;
        int idx = (h << 12) | loc;
        float2 a = S[loc];
        float p = a.x * a.x + a.y * a.y;
        #pragma unroll
        for (int i = 0; i < NQ; ++i)
            acc[i] += ((idx >> (15 - i)) & 1) ? -p : p;
    }
    __shared__ float red[8][NQ];
    #pragma unroll
    for (int i = 0; i < NQ; ++i) {
        float s = acc[i];
        for (int off = 16; off; off >>= 1) s += __shfl_xor(s, off, 32);
        if ((threadIdx.x & 31) == 0) red[threadIdx.x >> 5][i] = s;
    }
    __syncthreads();
    if (threadIdx.x < NQ) {
        float s = 0.f;
        #pragma unroll
        for (int wv = 0; wv < 8; ++wv) s += red[wv][threadIdx.x];  // fixed order
        part[((size_t)b * 16 + h) * NQ + threadIdx.x] = s;
    }
}

__global__ void __launch_bounds__(256) qc_reduce(const float* __restrict__ part,
                                                 float* __restrict__ out)
{
    int t = blockIdx.x * 256 + threadIdx.x;      // 8192 outputs
    if (t >= QBATCH * NQ) return;
    int b = t >> 4, q = t & 15;
    float s = 0.f;
    #pragma unroll
    for (int h = 0; h < 16; ++h)                 // fixed order -> deterministic
        s += part[((size_t)b * 16 + h) * NQ + q];
    out[t] = s;
}

// ---------------------------------------------------------------------------
extern "C" void kernel_launch(void* const* d_in, const int* in_sizes, int n_in,
                              void* d_out, int out_size, void* d_ws, size_t ws_size,
                              hipStream_t stream)
{
    const float* x      = (const float*)d_in[0];   // (512,16)
    const float* params = (const float*)d_in[1];   // (4,16,3)
    float* out = (float*)d_out;                    // (512,16)

    char* ws = (char*)d_ws;
    float*  gates = (float*)ws;                            // 64 KB
    float*  part  = (float*)(ws + (64 << 10));             // 512 KB
    float2* s0 = (float2*)(ws + (1 << 20));                // 256 MB
    float2* s1 = s0 + (size_t)QBATCH * NSTATE;             // 256 MB

    qc_build_gates<<<1, 16, 0, stream>>>(params, gates);
    qc_init_state<<<dim3(16, QBATCH), 256, 0, stream>>>(x, s0);

    float2* cur = s0;
    float2* nxt = s1;
    for (int L = 0; L < QDEPTH; ++L) {
        qc_passA<<<dim3(16, QBATCH), 256, 0, stream>>>(cur, gates + (size_t)(L * 4) * 1024);
        qc_passB<<<dim3(16, QBATCH), 256, 0, stream>>>(cur, nxt, gates + (size_t)(L * 4 + 3) * 1024);
        float2* tmp = cur; cur = nxt; nxt = tmp;
    }
    qc_expz<<<dim3(16, QBATCH), 256, 0, stream>>>(cur, part);
    qc_reduce<<<(QBATCH * NQ + 255) / 256, 256, 0, stream>>>(part, out);
}